// SpikingNeuronLayer_53523882443044
// MI455X (gfx1250) — compile-verified
//
#include <hip/hip_runtime.h>
#include <math.h>

typedef float v2f __attribute__((ext_vector_type(2)));
typedef float v4f __attribute__((ext_vector_type(4)));
typedef float v8f __attribute__((ext_vector_type(8)));

#define B_ 32
#define T_ 1024
#define D_ 512
#define H_ 512
#define M_ (B_ * T_)            // 32768 rows in the GEMM
#define ALPHA_ 0.951229424500714f  // exp(-1/20)

// ---------------------------------------------------------------------------
// Kernel 1: xlin[m][n] = sum_k x[m][k] * W[n][k] + b[n], fp32 via WMMA f32.
// One wave computes a 64x64 tile: 4x4 accumulator tiles (v8f each).
// Inner loop: 8 x b64 loads -> 16 x v_wmma_f32_16x16x4_f32.
// A frag (16x4 f32): lane l holds row M=l%16, K = (l/16)*2 + {0,1} -> float2.
// B frag (4x16 f32): lane l holds col N=l%16, K = (l/16)*2 + {0,1} -> float2
//   (B[k][n] = W[n][k], so same row-major float2 load pattern from W).
// C/D tile: VGPR r = row r (lanes 0-15) / row 8+r (lanes 16-31), N = lane%16.
// ---------------------------------------------------------------------------
__global__ __launch_bounds__(256) void snn_gemm_wmma(
    const float* __restrict__ x, const float* __restrict__ W,
    const float* __restrict__ bias, float* __restrict__ xlin)
{
    const int lane   = threadIdx.x & 31;
    const int wave   = threadIdx.x >> 5;
    const int waveId = blockIdx.x * 8 + wave;
    const int m64    = waveId & (M_ / 64 - 1);   // 512 m-blocks
    const int n64    = waveId / (M_ / 64);       // 8 n-blocks
    const int m0     = m64 * 64;
    const int n0     = n64 * 64;
    const int lm     = lane & 15;
    const int lk     = (lane >> 4) << 1;         // 0 or 2

    v8f c[4][4];
    #pragma unroll
    for (int i = 0; i < 4; ++i)
        #pragma unroll
        for (int j = 0; j < 4; ++j)
            c[i][j] = (v8f)(0.0f);

    const float* aBase = x + (size_t)(m0 + lm) * D_ + lk;
    const float* bBase = W + (size_t)(n0 + lm) * D_ + lk;

    for (int kk = 0; kk < D_; kk += 4) {
        v2f a[4], bf[4];
        #pragma unroll
        for (int i = 0; i < 4; ++i)
            a[i] = *(const v2f*)(aBase + (size_t)(i * 16) * D_ + kk);
        #pragma unroll
        for (int j = 0; j < 4; ++j)
            bf[j] = *(const v2f*)(bBase + (size_t)(j * 16) * D_ + kk);

        #pragma unroll
        for (int i = 0; i < 4; ++i)
            #pragma unroll
            for (int j = 0; j < 4; ++j)
                c[i][j] = __builtin_amdgcn_wmma_f32_16x16x4_f32(
                    /*neg_a=*/false, a[i], /*neg_b=*/false, bf[j],
                    /*c_mod=*/(short)0, c[i][j],
                    /*reuse_a=*/false, /*reuse_b=*/false);
    }

    float bv[4];
    #pragma unroll
    for (int j = 0; j < 4; ++j) bv[j] = bias[n0 + j * 16 + lm];

    const int rbase = (lane >> 4) * 8;
    #pragma unroll
    for (int i = 0; i < 4; ++i) {
        #pragma unroll
        for (int r = 0; r < 8; ++r) {
            const int row = m0 + i * 16 + rbase + r;
            float* outRow = xlin + (size_t)row * H_ + n0 + lm;
            #pragma unroll
            for (int j = 0; j < 4; ++j)
                outRow[j * 16] = c[i][j][r] + bv[j];
        }
    }
}

// ---------------------------------------------------------------------------
// Kernel 2: sequential LIF scan. One wave32 per batch row (32 blocks).
// Lane owns 16 hidden units: h = j*128 + lane*4 + {0..3} (coalesced v4f I/O).
// Per step: mem = alpha*mem + xlin[t]; butterfly-reduce sum & sumsq (no LDS,
// no barriers); std (ddof=1); threshold; reset; nontemporal spike store.
// Next-step loads are double-buffered to hide L2 latency behind the reduce.
// ---------------------------------------------------------------------------
__global__ __launch_bounds__(32) void snn_scan(
    const float* __restrict__ xlin, const float* __restrict__ thr_adapt,
    float* __restrict__ spikes, float* __restrict__ mem_out)
{
    const int b    = blockIdx.x;
    const int lane = threadIdx.x;

    float mem[16];
    float thr0[16];
    #pragma unroll
    for (int k = 0; k < 16; ++k) mem[k] = 0.0f;

    #pragma unroll
    for (int j = 0; j < 4; ++j) {
        v4f tv = *(const v4f*)(thr_adapt + j * 128 + lane * 4);
        thr0[j * 4 + 0] = tv.x; thr0[j * 4 + 1] = tv.y;
        thr0[j * 4 + 2] = tv.z; thr0[j * 4 + 3] = tv.w;
    }

    const float* xb = xlin + (size_t)b * T_ * H_;

    v4f cur[4];
    #pragma unroll
    for (int j = 0; j < 4; ++j)
        cur[j] = *(const v4f*)(xb + j * 128 + lane * 4);

    for (int t = 0; t < T_; ++t) {
        v4f nxt[4];
        if (t + 1 < T_) {
            const float* xn = xb + (size_t)(t + 1) * H_;
            #pragma unroll
            for (int j = 0; j < 4; ++j)
                nxt[j] = *(const v4f*)(xn + j * 128 + lane * 4);
        }

        float s = 0.0f, ss = 0.0f;
        #pragma unroll
        for (int j = 0; j < 4; ++j) {
            #pragma unroll
            for (int cc = 0; cc < 4; ++cc) {
                const int k = j * 4 + cc;
                float m = fmaf(ALPHA_, mem[k], cur[j][cc]);
                mem[k] = m;
                s  += m;
                ss  = fmaf(m, m, ss);
            }
        }

        // wave32 butterfly reduction (sum and sum-of-squares)
        #pragma unroll
        for (int off = 16; off > 0; off >>= 1) {
            s  += __shfl_xor(s,  off, 32);
            ss += __shfl_xor(ss, off, 32);
        }

        // unbiased std over H: var = (ss - s^2/H) / (H-1)
        const float var = (ss - s * s * (1.0f / (float)H_)) * (1.0f / (float)(H_ - 1));
        const float sd  = sqrtf(fmaxf(var, 0.0f));
        const float add = 0.1f * sd;

        float* sout = spikes + ((size_t)b * T_ + t) * H_;
        #pragma unroll
        for (int j = 0; j < 4; ++j) {
            v4f sp;
            #pragma unroll
            for (int cc = 0; cc < 4; ++cc) {
                const int k = j * 4 + cc;
                const float spk = (mem[k] >= thr0[k] + add) ? 1.0f : 0.0f;
                mem[k] = (spk != 0.0f) ? 0.0f : mem[k];
                sp[cc] = spk;
            }
            // write-once stream: keep it out of L2 so xlin stays resident
            __builtin_nontemporal_store(sp, (v4f*)(sout + j * 128 + lane * 4));
        }

        #pragma unroll
        for (int j = 0; j < 4; ++j) cur[j] = nxt[j];
    }

    float* mo = mem_out + (size_t)b * H_;
    #pragma unroll
    for (int j = 0; j < 4; ++j) {
        v4f mv;
        #pragma unroll
        for (int cc = 0; cc < 4; ++cc) mv[cc] = mem[j * 4 + cc];
        *(v4f*)(mo + j * 128 + lane * 4) = mv;
    }
}

// ---------------------------------------------------------------------------
extern "C" void kernel_launch(void* const* d_in, const int* in_sizes, int n_in,
                              void* d_out, int out_size, void* d_ws, size_t ws_size,
                              hipStream_t stream)
{
    (void)in_sizes; (void)n_in; (void)out_size; (void)ws_size;

    const float* x    = (const float*)d_in[0];   // [B,T,D]
    const float* W    = (const float*)d_in[1];   // [H,D]
    const float* bias = (const float*)d_in[2];   // [H]
    const float* thr  = (const float*)d_in[3];   // [H]

    float* out     = (float*)d_out;
    float* spikes  = out;                               // [B,T,H]
    float* mem_out = out + (size_t)B_ * T_ * H_;        // [B,H]
    float* xlin    = (float*)d_ws;                      // [B*T,H] = 64 MB

    // GEMM: (M/64)*(H/64) = 512*8 = 4096 waves, 8 waves/block -> 512 blocks
    snn_gemm_wmma<<<dim3((M_ / 64) * (H_ / 64) / 8), dim3(256), 0, stream>>>(
        x, W, bias, xlin);

    // Scan: one wave32 per batch row
    snn_scan<<<dim3(B_), dim3(32), 0, stream>>>(xlin, thr, spikes, mem_out);
}